// GATv2ConvolutionLinSkip_72911364817019
// MI455X (gfx1250) — compile-verified
//
#include <hip/hip_runtime.h>
#include <math.h>

#define NSLOPE 0.2f
#define HDIM 64
#define CDIM 40
#define FIN1 128

typedef __attribute__((ext_vector_type(2))) float v2f;
typedef __attribute__((ext_vector_type(8))) float v8f;

// ---------------------------------------------------------------------------
// Dual GEMM with fp32 WMMA:  XL = X*Wl + bl,  XR = X*Wr + br
// X: [N, FIN], Wl/Wr: [FIN, 64].  One wave -> 16x16 output tile, shared A frag.
// A layout (32-bit 16x4): lanes 0-15 hold K=0,1 in v0,v1; lanes 16-31 K=2,3.
// B layout (4x16): v0: K=0 (lanes 0-15) / K=2 (lanes 16-31); v1: K=1 / K=3.
// C/D layout: VGPR v -> row m0+v (lanes 0-15), m0+8+v (lanes 16-31).
// ---------------------------------------------------------------------------
template <int FIN>
__global__ __launch_bounds__(32) void gemm_dual_wmma(
    const float* __restrict__ X, const float* __restrict__ Wl,
    const float* __restrict__ Wr, const float* __restrict__ bl,
    const float* __restrict__ br, float* __restrict__ XL,
    float* __restrict__ XR, int Nn) {
  const int m0 = blockIdx.x * 16;
  const int n0 = blockIdx.y * 16;
  const int lane = threadIdx.x;
  const int row = lane & 15;
  const int kh = (lane >> 4) * 2;  // 0 or 2

  int mr = m0 + row;
  if (mr >= Nn) mr = Nn - 1;  // clamp (EXEC must stay all-ones for WMMA)
  const float* xrow = X + (size_t)mr * FIN;

  v8f accL = {};
  v8f accR = {};
  for (int k = 0; k < FIN; k += 4) {
    v2f a, bL, bR;
    a.x = xrow[k + kh];
    a.y = xrow[k + kh + 1];
    bL.x = Wl[(size_t)(k + kh) * HDIM + n0 + row];
    bL.y = Wl[(size_t)(k + kh + 1) * HDIM + n0 + row];
    bR.x = Wr[(size_t)(k + kh) * HDIM + n0 + row];
    bR.y = Wr[(size_t)(k + kh + 1) * HDIM + n0 + row];
    accL = __builtin_amdgcn_wmma_f32_16x16x4_f32(false, a, false, bL,
                                                 (short)0, accL, false, false);
    accR = __builtin_amdgcn_wmma_f32_16x16x4_f32(false, a, false, bR,
                                                 (short)0, accR, false, false);
  }

  const float blv = bl[n0 + row];
  const float brv = br[n0 + row];
  const int mbase = m0 + (lane >> 4) * 8;
#pragma unroll
  for (int v = 0; v < 8; ++v) {
    const int m = mbase + v;
    if (m < Nn) {
      XL[(size_t)m * HDIM + n0 + row] = accL[v] + blv;
      XR[(size_t)m * HDIM + n0 + row] = accR[v] + brv;
    }
  }
}

// ---------------------------------------------------------------------------
__global__ void reset_kernel(float* __restrict__ agg, int* __restrict__ mbits,
                             float* __restrict__ denom, int Nn) {
  const int i = blockIdx.x * blockDim.x + threadIdx.x;
  const int NH = Nn * HDIM;
  if (i < NH) agg[i] = 0.0f;
  if (i < Nn) {
    mbits[i] = (int)0xFF800000u;  // -inf
    denom[i] = 0.0f;
  }
}

__device__ __forceinline__ void atomic_max_float_bits(int* addr, float val) {
  const int iv = __float_as_int(val);
  if (iv >= 0)
    atomicMax(addr, iv);
  else
    atomicMin((unsigned int*)addr, (unsigned int)iv);
}

__device__ __forceinline__ float lrelu(float v) {
  return v > 0.0f ? v : NSLOPE * v;
}

// e = att . LeakyReLU(xl[src] + xr[dst]);  segment-max into mbits[dst]
__global__ void edge_score(const float* __restrict__ xl,
                           const float* __restrict__ xr,
                           const float* __restrict__ att,
                           const int* __restrict__ eidx, int E, int EP,
                           float* __restrict__ e_sc, int* __restrict__ mbits) {
  __shared__ float s_att[HDIM];
  if (threadIdx.x < HDIM) s_att[threadIdx.x] = att[threadIdx.x];
  __syncthreads();
  const int e = blockIdx.x * blockDim.x + threadIdx.x;
  if (e >= EP) return;
  const int s = (e < E) ? eidx[e] : (e - E);
  const int d = (e < E) ? eidx[E + e] : (e - E);
  const float4* pl = (const float4*)(xl + (size_t)s * HDIM);
  const float4* pr = (const float4*)(xr + (size_t)d * HDIM);
  float acc = 0.0f;
#pragma unroll 4
  for (int i = 0; i < HDIM / 4; ++i) {
    const float4 a = pl[i];
    const float4 b = pr[i];
    acc += s_att[4 * i + 0] * lrelu(a.x + b.x);
    acc += s_att[4 * i + 1] * lrelu(a.y + b.y);
    acc += s_att[4 * i + 2] * lrelu(a.z + b.z);
    acc += s_att[4 * i + 3] * lrelu(a.w + b.w);
  }
  e_sc[e] = acc;
  atomic_max_float_bits(mbits + d, acc);
}

// p = exp(e - m[dst]);  segment-sum into denom[dst]
__global__ void edge_exp_sum(const float* __restrict__ e_sc,
                             const int* __restrict__ mbits,
                             const int* __restrict__ eidx, int E, int EP,
                             float* __restrict__ p_sc,
                             float* __restrict__ denom) {
  const int e = blockIdx.x * blockDim.x + threadIdx.x;
  if (e >= EP) return;
  const int d = (e < E) ? eidx[E + e] : (e - E);
  const float p = __expf(e_sc[e] - __int_as_float(mbits[d]));
  p_sc[e] = p;
  atomicAdd(denom + d, p);
}

// agg[dst] += (p/denom[dst]) * xl[src]   (64 lanes per edge, coalesced)
__global__ void edge_aggregate(const float* __restrict__ xl,
                               const float* __restrict__ p_sc,
                               const float* __restrict__ denom,
                               const int* __restrict__ eidx, int E, int EP,
                               float* __restrict__ agg) {
  const long long t = (long long)blockIdx.x * blockDim.x + threadIdx.x;
  const int e = (int)(t >> 6);
  const int h = (int)(t & 63);
  if (e >= EP) return;
  const int s = (e < E) ? eidx[e] : (e - E);
  const int d = (e < E) ? eidx[E + e] : (e - E);
  const float alpha = p_sc[e] / (denom[d] + 1e-16f);
  atomicAdd(&agg[(size_t)d * HDIM + h], alpha * xl[(size_t)s * HDIM + h]);
}

__global__ void finish_relu(const float* __restrict__ agg,
                            const float* __restrict__ bias,
                            float* __restrict__ out, int NH) {
  const int i = blockIdx.x * blockDim.x + threadIdx.x;
  if (i >= NH) return;
  const float v = agg[i] + bias[i & (HDIM - 1)];
  out[i] = v > 0.0f ? v : 0.0f;
}

__global__ void finish_residual(const float* __restrict__ h1,
                                const float* __restrict__ agg,
                                const float* __restrict__ bias,
                                float* __restrict__ out, int NH) {
  const int i = blockIdx.x * blockDim.x + threadIdx.x;
  if (i >= NH) return;
  out[i] = h1[i] + agg[i] + bias[i & (HDIM - 1)];
}

// logits = h @ Wlin + blin; log_softmax over C=40 (thread per node)
__global__ __launch_bounds__(128) void head_logsoftmax(
    const float* __restrict__ hres, const float* __restrict__ Wlin,
    const float* __restrict__ blin, float* __restrict__ out, int Nn) {
  const int n = blockIdx.x * blockDim.x + threadIdx.x;
  if (n >= Nn) return;
  float hv[HDIM];
  const float4* hp = (const float4*)(hres + (size_t)n * HDIM);
#pragma unroll
  for (int i = 0; i < HDIM / 4; ++i) {
    const float4 v = hp[i];
    hv[4 * i + 0] = v.x;
    hv[4 * i + 1] = v.y;
    hv[4 * i + 2] = v.z;
    hv[4 * i + 3] = v.w;
  }
  float logits[CDIM];
  float mx = -INFINITY;
  for (int c = 0; c < CDIM; ++c) {
    float acc = blin[c];
#pragma unroll 8
    for (int h = 0; h < HDIM; ++h) acc += hv[h] * Wlin[h * CDIM + c];
    logits[c] = acc;
    mx = fmaxf(mx, acc);
  }
  float se = 0.0f;
  for (int c = 0; c < CDIM; ++c) se += __expf(logits[c] - mx);
  const float lse = mx + __logf(se);
  float* op = out + (size_t)n * CDIM;
  for (int c = 0; c < CDIM; ++c) op[c] = logits[c] - lse;
}

// ---------------------------------------------------------------------------
extern "C" void kernel_launch(void* const* d_in, const int* in_sizes, int n_in,
                              void* d_out, int out_size, void* d_ws,
                              size_t ws_size, hipStream_t stream) {
  const float* x = (const float*)d_in[0];
  const int* eidx = (const int*)d_in[1];
  const float* W1l = (const float*)d_in[2];
  const float* W1r = (const float*)d_in[3];
  const float* bl1 = (const float*)d_in[4];
  const float* br1 = (const float*)d_in[5];
  const float* att1 = (const float*)d_in[6];
  const float* bias1 = (const float*)d_in[7];
  const float* W2l = (const float*)d_in[8];
  const float* W2r = (const float*)d_in[9];
  const float* bl2 = (const float*)d_in[10];
  const float* br2 = (const float*)d_in[11];
  const float* att2 = (const float*)d_in[12];
  const float* bias2 = (const float*)d_in[13];
  const float* Wlin = (const float*)d_in[14];
  const float* blin = (const float*)d_in[15];
  float* out = (float*)d_out;

  const int Nn = in_sizes[0] / FIN1;
  const int E = in_sizes[1] / 2;
  const int EP = E + Nn;  // with self-loops
  const int NH = Nn * HDIM;

  float* ws = (float*)d_ws;
  float* xl = ws;
  float* xr = xl + (size_t)NH;
  float* agg = xr + (size_t)NH;
  float* h1 = agg + (size_t)NH;
  float* hres = h1 + (size_t)NH;
  float* e_sc = hres + (size_t)NH;
  float* p_sc = e_sc + (size_t)EP;
  float* mval = p_sc + (size_t)EP;  // float bits, used as int*
  float* denom = mval + (size_t)Nn;
  int* mbits = (int*)mval;

  const dim3 gemm_grid((Nn + 15) / 16, HDIM / 16);
  const int eb = (EP + 255) / 256;
  const int nhb = (NH + 255) / 256;
  const long long aggT = (long long)EP * HDIM;
  const int aggb = (int)((aggT + 255) / 256);

  // ---- layer 1 ----
  gemm_dual_wmma<FIN1><<<gemm_grid, 32, 0, stream>>>(x, W1l, W1r, bl1, br1, xl,
                                                     xr, Nn);
  reset_kernel<<<nhb, 256, 0, stream>>>(agg, mbits, denom, Nn);
  edge_score<<<eb, 256, 0, stream>>>(xl, xr, att1, eidx, E, EP, e_sc, mbits);
  edge_exp_sum<<<eb, 256, 0, stream>>>(e_sc, mbits, eidx, E, EP, p_sc, denom);
  edge_aggregate<<<aggb, 256, 0, stream>>>(xl, p_sc, denom, eidx, E, EP, agg);
  finish_relu<<<nhb, 256, 0, stream>>>(agg, bias1, h1, NH);

  // ---- layer 2 ----
  gemm_dual_wmma<HDIM><<<gemm_grid, 32, 0, stream>>>(h1, W2l, W2r, bl2, br2,
                                                     xl, xr, Nn);
  reset_kernel<<<nhb, 256, 0, stream>>>(agg, mbits, denom, Nn);
  edge_score<<<eb, 256, 0, stream>>>(xl, xr, att2, eidx, E, EP, e_sc, mbits);
  edge_exp_sum<<<eb, 256, 0, stream>>>(e_sc, mbits, eidx, E, EP, p_sc, denom);
  edge_aggregate<<<aggb, 256, 0, stream>>>(xl, p_sc, denom, eidx, E, EP, agg);
  finish_residual<<<nhb, 256, 0, stream>>>(h1, agg, bias2, hres, NH);

  // ---- head ----
  head_logsoftmax<<<(Nn + 127) / 128, 128, 0, stream>>>(hres, Wlin, blin, out,
                                                        Nn);
}